// PCTEncoder_61332132987238
// MI455X (gfx1250) — compile-verified
//
#include <hip/hip_runtime.h>
#include <math.h>

typedef __attribute__((ext_vector_type(2))) float v2f;
typedef __attribute__((ext_vector_type(8))) float v8f;

#define NB 4
#define NPTS 2048
#define KNN 32
#define C1 128
#define C2 256
#define DQ 32
#define CG 1024
#define BN_EPS 1e-5f
#define SLOPE 0.01f

static __device__ __forceinline__ v8f wmma4(v2f a, v2f b, v8f c) {
  return __builtin_amdgcn_wmma_f32_16x16x4_f32(false, a, false, b, (short)0, c, false, false);
}

// ---------------- embed: x[3] -> 128 -> 128 with BN+LeakyReLU ----------------
__global__ void k_embed(const float* __restrict__ x,
                        const float* __restrict__ W1, const float* __restrict__ g1,
                        const float* __restrict__ b1, const float* __restrict__ m1,
                        const float* __restrict__ v1,
                        const float* __restrict__ W2, const float* __restrict__ g2,
                        const float* __restrict__ b2, const float* __restrict__ m2,
                        const float* __restrict__ v2,
                        float* __restrict__ h1) {
  __shared__ float tl[C1];
  int p = blockIdx.x, o = threadIdx.x;
  float x0 = x[p * 3 + 0], x1 = x[p * 3 + 1], x2 = x[p * 3 + 2];
  float y = W1[o * 3 + 0] * x0 + W1[o * 3 + 1] * x1 + W1[o * 3 + 2] * x2;
  float s = g1[o] * rsqrtf(v1[o] + BN_EPS);
  y = (y - m1[o]) * s + b1[o];
  tl[o] = y >= 0.f ? y : SLOPE * y;
  __syncthreads();
  float acc = 0.f;
  const float* wr = W2 + (size_t)o * C1;
  for (int j = 0; j < C1; ++j) acc += tl[j] * wr[j];
  s = g2[o] * rsqrtf(v2[o] + BN_EPS);
  acc = (acc - m2[o]) * s + b2[o];
  h1[(size_t)p * C1 + o] = acc >= 0.f ? acc : SLOPE * acc;
}

// ---------------- squared norms ----------------
__global__ void k_sqnorm(const float* __restrict__ h1, float* __restrict__ sqn) {
  int p = blockIdx.x * blockDim.x + threadIdx.x;
  const float* r = h1 + (size_t)p * C1;
  float a = 0.f;
  for (int c = 0; c < C1; ++c) a += r[c] * r[c];
  sqn[p] = a;
}

// ---------------- kNN: 4 waves/block, one query per wave ----------------
__global__ void k_knn(const float* __restrict__ h1, const float* __restrict__ sqn,
                      int* __restrict__ knn) {
  __shared__ float dsm[4][NPTS];
  __shared__ float hq[4][C1];
  int lane = threadIdx.x & 31, w = threadIdx.x >> 5;
  int p = blockIdx.x * 4 + w;
  int bbase = (p >> 11) << 11;
  for (int c = lane; c < C1; c += 32) hq[w][c] = h1[(size_t)p * C1 + c];
  __syncthreads();
  float sqp = sqn[p];
  for (int i = 0; i < NPTS / 32; ++i) {
    int m = lane + (i << 5);
    const float* hm = h1 + (size_t)(bbase + m) * C1;
    float acc = 0.f;
    for (int c = 0; c < C1; ++c) acc += hq[w][c] * hm[c];
    dsm[w][m] = sqp + sqn[bbase + m] - 2.f * acc;
  }
  __syncthreads();
  for (int r = 0; r < KNN; ++r) {
    float lv = 3.4e38f;
    int li = 0x7fffffff;
    for (int i = 0; i < NPTS / 32; ++i) {
      int m = lane + (i << 5);
      float v = dsm[w][m];
      if (v < lv) { lv = v; li = m; }
    }
    for (int off = 16; off; off >>= 1) {
      float ov = __shfl_xor(lv, off, 32);
      int oi = __shfl_xor(li, off, 32);
      if (ov < lv || (ov == lv && oi < li)) { lv = ov; li = oi; }
    }
    if (lane == 0) knn[(size_t)p * KNN + r] = bbase + li;
    if (lane == (li & 31)) dsm[w][li] = 3.4e38f;
    __syncthreads();
  }
}

// ---------------- edge conv: per point, E[32,256] -> GEMM -> GEMM -> max ----------------
static __device__ __forceinline__ void store_act_cm(float T[C2][KNN], int mbase, int nbase,
                                                    v8f acc, const float* g, const float* bb,
                                                    const float* mu, const float* vv, int lane) {
  int col = nbase + (lane & 15);
  int rhi = (lane >> 4) * 8;
  float s = g[col] * rsqrtf(vv[col] + BN_EPS);
  float sh = bb[col] - mu[col] * s;
#pragma unroll
  for (int r = 0; r < 8; ++r) {
    float val = acc[r] * s + sh;
    T[col][mbase + r + rhi] = val >= 0.f ? val : SLOPE * val;
  }
}

__global__ void k_edgeconv(const float* __restrict__ h1, const int* __restrict__ knn,
                           const float* __restrict__ Wa, const float* __restrict__ ga,
                           const float* __restrict__ ba, const float* __restrict__ ma,
                           const float* __restrict__ va,
                           const float* __restrict__ Wb, const float* __restrict__ gb,
                           const float* __restrict__ bb, const float* __restrict__ mb,
                           const float* __restrict__ vb,
                           float* __restrict__ gmax) {
  // column-major tiles: [channel][neighbor], exactly 64 KB total
  __shared__ float Es[C2][KNN];
  __shared__ float Ts[C2][KNN];
  int p = blockIdx.x;
  int t = threadIdx.x;
  const float* ctr = h1 + (size_t)p * C1;
  for (int i = t; i < C2 * KNN; i += 256) {
    int c = i >> 5, j = i & 31;
    if (c < C1) {
      int nb = knn[(size_t)p * KNN + j];
      Es[c][j] = h1[(size_t)nb * C1 + c] - ctr[c];
    } else {
      Es[c][j] = ctr[c - C1];
    }
  }
  __syncthreads();
  int lane = t & 31, w = t >> 5;
  int nb0 = w * 32;  // this wave covers output columns [nb0, nb0+32)
  int am = lane & 15, k0 = (lane >> 4) * 2;
  // GEMM1: Es x Wa^T -> Ts
  {
    v8f a00 = {}, a01 = {}, a10 = {}, a11 = {};
    for (int kb = 0; kb < C2; kb += 4) {
      v2f a0, a1, b0, b1;
      a0.x = Es[kb + k0][am];      a0.y = Es[kb + k0 + 1][am];
      a1.x = Es[kb + k0][am + 16]; a1.y = Es[kb + k0 + 1][am + 16];
      const float* w0 = Wa + (size_t)(nb0 + am) * C2 + kb + k0;
      const float* w1 = Wa + (size_t)(nb0 + 16 + am) * C2 + kb + k0;
      b0.x = w0[0]; b0.y = w0[1];
      b1.x = w1[0]; b1.y = w1[1];
      a00 = wmma4(a0, b0, a00);
      a01 = wmma4(a0, b1, a01);
      a10 = wmma4(a1, b0, a10);
      a11 = wmma4(a1, b1, a11);
    }
    store_act_cm(Ts, 0, nb0, a00, ga, ba, ma, va, lane);
    store_act_cm(Ts, 0, nb0 + 16, a01, ga, ba, ma, va, lane);
    store_act_cm(Ts, 16, nb0, a10, ga, ba, ma, va, lane);
    store_act_cm(Ts, 16, nb0 + 16, a11, ga, ba, ma, va, lane);
  }
  __syncthreads();
  // GEMM2: Ts x Wb^T -> Es (reuse)
  {
    v8f a00 = {}, a01 = {}, a10 = {}, a11 = {};
    for (int kb = 0; kb < C2; kb += 4) {
      v2f a0, a1, b0, b1;
      a0.x = Ts[kb + k0][am];      a0.y = Ts[kb + k0 + 1][am];
      a1.x = Ts[kb + k0][am + 16]; a1.y = Ts[kb + k0 + 1][am + 16];
      const float* w0 = Wb + (size_t)(nb0 + am) * C2 + kb + k0;
      const float* w1 = Wb + (size_t)(nb0 + 16 + am) * C2 + kb + k0;
      b0.x = w0[0]; b0.y = w0[1];
      b1.x = w1[0]; b1.y = w1[1];
      a00 = wmma4(a0, b0, a00);
      a01 = wmma4(a0, b1, a01);
      a10 = wmma4(a1, b0, a10);
      a11 = wmma4(a1, b1, a11);
    }
    store_act_cm(Es, 0, nb0, a00, gb, bb, mb, vb, lane);
    store_act_cm(Es, 0, nb0 + 16, a01, gb, bb, mb, vb, lane);
    store_act_cm(Es, 16, nb0, a10, gb, bb, mb, vb, lane);
    store_act_cm(Es, 16, nb0 + 16, a11, gb, bb, mb, vb, lane);
  }
  __syncthreads();
  // max over 32 neighbors, thread t = channel
  float mx = -3.4e38f;
  for (int j = 0; j < KNN; ++j) mx = fmaxf(mx, Es[t][j]);
  gmax[(size_t)p * C2 + t] = mx;
}

// ---------------- generic WMMA GEMM: C = act(A * W^T) [+ res] ----------------
// A: rows indexed by grid.x*32, A[row*lda + acol + k]; W: [Ncols, K] row-major.
// fuse bit0: BN+LeakyReLU, bit1: add residual.
__global__ void k_gemm(const float* __restrict__ A, int lda, int acol,
                       const float* __restrict__ W, int K, int Ncols,
                       const float* __restrict__ bg, const float* __restrict__ bbta,
                       const float* __restrict__ bmu, const float* __restrict__ bvar,
                       const float* __restrict__ res, int ldr, int rcol,
                       float* __restrict__ C, int ldc, int ccol, int fuse) {
  int lane = threadIdx.x & 31;
  int w = threadIdx.x >> 5;
  int mt = w & 1, nt = w >> 1;
  int m_base = blockIdx.x * 32 + mt * 16;
  int n_base = blockIdx.y * 64 + nt * 16;
  if (n_base >= Ncols) return;
  int am = lane & 15, k0 = (lane >> 4) * 2;
  const float* arow = A + (size_t)(m_base + am) * lda + acol;
  const float* wrow = W + (size_t)(n_base + am) * K;
  v8f acc = {};
  for (int kb = 0; kb < K; kb += 4) {
    v2f a, b;
    a.x = arow[kb + k0]; a.y = arow[kb + k0 + 1];
    b.x = wrow[kb + k0]; b.y = wrow[kb + k0 + 1];
    acc = wmma4(a, b, acc);
  }
  int rhi = (lane >> 4) * 8;
  int col = n_base + (lane & 15);
  float s = 1.f, sh = 0.f;
  if (fuse & 1) {
    s = bg[col] * rsqrtf(bvar[col] + BN_EPS);
    sh = bbta[col] - bmu[col] * s;
  }
#pragma unroll
  for (int r = 0; r < 8; ++r) {
    int row = m_base + r + rhi;
    float val = acc[r];
    if (fuse & 1) { val = val * s + sh; val = val >= 0.f ? val : SLOPE * val; }
    if (fuse & 2) val += res[(size_t)row * ldr + rcol + col];
    C[(size_t)row * ldc + ccol + col] = val;
  }
}

// ---------------- attention pass 1: per-column online max / sumexp ----------------
__global__ void k_colstats(const float* __restrict__ q, const float* __restrict__ k,
                           float* __restrict__ cmax, float* __restrict__ csum) {
  int idx = blockIdx.x * 256 + threadIdx.x;  // b*N + m
  int bbase = (idx >> 11) << 11;
  float kr[DQ];
#pragma unroll
  for (int c = 0; c < DQ; ++c) kr[c] = k[(size_t)idx * DQ + c];
  float mx = -3.4e38f, s = 0.f;
  for (int n = 0; n < NPTS; ++n) {
    const float* qr = q + (size_t)(bbase + n) * DQ;
    float e = 0.f;
#pragma unroll
    for (int c = 0; c < DQ; ++c) e += qr[c] * kr[c];
    float nm = fmaxf(mx, e);
    s = s * __expf(mx - nm) + __expf(e - nm);
    mx = nm;
  }
  cmax[idx] = mx;
  csum[idx] = s;
}

// ---------------- attention pass 2: row weights, L1 renorm, apply V, diff ----------------
__global__ void k_rowapply(const float* __restrict__ q, const float* __restrict__ k,
                           const float* __restrict__ v, const float* __restrict__ cmax,
                           const float* __restrict__ csum,
                           const float* __restrict__ X, int ldx, int xcol,
                           float* __restrict__ diff) {
  __shared__ float qs[DQ];
  __shared__ float wm[NPTS];
  __shared__ float red[256];
  int p = blockIdx.x, t = threadIdx.x;
  int bbase = (p >> 11) << 11;
  if (t < DQ) qs[t] = q[(size_t)p * DQ + t];
  __syncthreads();
  float part = 0.f;
  for (int m = t; m < NPTS; m += 256) {
    const float* kr = k + (size_t)(bbase + m) * DQ;
    float e = 0.f;
#pragma unroll
    for (int c = 0; c < DQ; ++c) e += qs[c] * kr[c];
    float a = __expf(e - cmax[bbase + m]) / csum[bbase + m];
    wm[m] = a;
    part += a;
  }
  red[t] = part;
  __syncthreads();
  for (int off = 128; off; off >>= 1) {
    if (t < off) red[t] += red[t + off];
    __syncthreads();
  }
  float inv = 1.f / (1e-9f + red[0]);
  float xr = 0.f;
  for (int m = 0; m < NPTS; ++m) xr += wm[m] * v[(size_t)(bbase + m) * C2 + t];
  diff[(size_t)p * C2 + t] = X[(size_t)p * ldx + xcol + t] - inv * xr;
}

// ---------------- host launch ----------------
extern "C" void kernel_launch(void* const* d_in, const int* in_sizes, int n_in,
                              void* d_out, int out_size, void* d_ws, size_t ws_size,
                              hipStream_t stream) {
  (void)in_sizes; (void)n_in; (void)out_size; (void)ws_size;
  const int M = NB * NPTS;  // 8192
  auto F = [&](int i) { return (const float*)d_in[i]; };

  const float* x = F(0);
  // e1a:1-5 e1b:6-10 e2a:11-15 e2b:16-20 oa_i base 21+8i (Wq,Wk,Wv,pW,pg,pb,pm,pv) lbr1:53-57
  float* h1   = (float*)d_ws;                   // M*128
  float* sqn  = h1 + (size_t)M * C1;            // M
  int*   knn  = (int*)(sqn + M);                // M*32
  float* gmax = (float*)(knn + (size_t)M * KNN);// M*256
  float* Obuf = gmax + (size_t)M * C2;          // M*1024 (o1..o4 concatenated)
  float* qb   = Obuf + (size_t)M * CG;          // M*32
  float* kb   = qb + (size_t)M * DQ;            // M*32
  float* vb   = kb + (size_t)M * DQ;            // M*256
  float* cmax = vb + (size_t)M * C2;            // M
  float* csum = cmax + M;                       // M
  float* diff = csum + M;                       // M*256
  float* out  = (float*)d_out;

  k_embed<<<M, C1, 0, stream>>>(x, F(1), F(2), F(3), F(4), F(5),
                                F(6), F(7), F(8), F(9), F(10), h1);
  k_sqnorm<<<M / 256, 256, 0, stream>>>(h1, sqn);
  k_knn<<<M / 4, 128, 0, stream>>>(h1, sqn, knn);
  k_edgeconv<<<M, 256, 0, stream>>>(h1, knn,
                                    F(11), F(12), F(13), F(14), F(15),
                                    F(16), F(17), F(18), F(19), F(20), gmax);

  for (int i = 0; i < 4; ++i) {
    int base = 21 + 8 * i;
    const float* Wq = F(base + 0);
    const float* Wk = F(base + 1);
    const float* Wv = F(base + 2);
    const float* pW = F(base + 3);
    const float* pg = F(base + 4);
    const float* pb = F(base + 5);
    const float* pm = F(base + 6);
    const float* pv = F(base + 7);
    const float* Xp = (i == 0) ? gmax : Obuf;
    int ldx = (i == 0) ? C2 : CG;
    int xcol = (i == 0) ? 0 : (i - 1) * C2;

    k_gemm<<<dim3(M / 32, 1), 256, 0, stream>>>(Xp, ldx, xcol, Wq, C2, DQ,
        nullptr, nullptr, nullptr, nullptr, nullptr, 0, 0, qb, DQ, 0, 0);
    k_gemm<<<dim3(M / 32, 1), 256, 0, stream>>>(Xp, ldx, xcol, Wk, C2, DQ,
        nullptr, nullptr, nullptr, nullptr, nullptr, 0, 0, kb, DQ, 0, 0);
    k_gemm<<<dim3(M / 32, 4), 256, 0, stream>>>(Xp, ldx, xcol, Wv, C2, C2,
        nullptr, nullptr, nullptr, nullptr, nullptr, 0, 0, vb, C2, 0, 0);
    k_colstats<<<M / 256, 256, 0, stream>>>(qb, kb, cmax, csum);
    k_rowapply<<<M, 256, 0, stream>>>(qb, kb, vb, cmax, csum, Xp, ldx, xcol, diff);
    // o_i = lbrd(diff, proj) + X, written into Obuf column slice i*256
    k_gemm<<<dim3(M / 32, 4), 256, 0, stream>>>(diff, C2, 0, pW, C2, C2,
        pg, pb, pm, pv, Xp, ldx, xcol, Obuf, CG, i * C2, 3);
  }

  // lbr1: [8192,1024] x [1024,1024]^T with BN+LeakyReLU -> d_out
  k_gemm<<<dim3(M / 32, CG / 64), 256, 0, stream>>>(Obuf, CG, 0, F(53), CG, CG,
      F(54), F(55), F(56), F(57), nullptr, 0, 0, out, CG, 0, 1);
}